// Focal_loss_sigmoid_44040594653646
// MI455X (gfx1250) — compile-verified
//
#include <hip/hip_runtime.h>
#include <math.h>
#include <stdint.h>

// Flip these to 0 if a toolchain rejects the corresponding gfx1250 path.
#ifndef USE_ASYNC_LDS
#define USE_ASYNC_LDS 1
#endif
#ifndef USE_TDM
#define USE_TDM 1
#endif

#if USE_TDM && defined(__has_builtin)
#if __has_builtin(__builtin_amdgcn_tensor_load_to_lds)
#define TDM_OK 1
#endif
#endif
#ifndef TDM_OK
#define TDM_OK 0
#endif

typedef unsigned int v4u __attribute__((ext_vector_type(4)));
typedef int v4i __attribute__((ext_vector_type(4)));
typedef int v8i __attribute__((ext_vector_type(8)));

namespace {
constexpr int kB = 2, kA = 3, kNCLS = 3, kC = kNCLS * kA;  // 9 channels
constexpr int kD = 64, kH = 128, kW = 128, kM = 128;
constexpr float kEPS = 1e-4f;
// x > ln(EPS/(1-EPS))  <=>  sigmoid(x) > EPS  (exact monotone rewrite)
constexpr float kXThresh = -9.21024036697585f;
constexpr int kTH = 8;                       // H rows per block
constexpr int kHT = kH / kTH;                // 16 tiles along H
constexpr int kNBLK = kB * kC * kD * kHT;    // 18432 blocks
constexpr int kPlane = 10 * kW;              // 1280 floats (8 rows + 2 halo)
constexpr int kTile = 3 * kPlane;            // 3840 floats (3 d-planes)
constexpr int kVec = kTile / 4;              // 960 16-byte chunks
constexpr int kVol = kD * kH * kW;           // per-channel voxel count
}  // namespace

// ---------------------------------------------------------------------------
// Kernel 1: negative-branch focal loss with 3x3x3 NMS max-pool.
// Interior blocks stage their 3x10x128 halo tile with ONE Tensor-Data-Mover
// descriptor (tensor_load_to_lds, TENSORcnt); boundary blocks use per-lane
// async global->LDS b128 DMA with -inf halo fill (ASYNCcnt). Separable max
// in LDS, mask-gated transcendentals, deterministic block reduction.
// ---------------------------------------------------------------------------
__global__ __launch_bounds__(256) void focal_neg_kernel(
    const float* __restrict__ logits, const float* __restrict__ prob_gt,
    float* __restrict__ ws_loss, float* __restrict__ ws_cnt) {
  __shared__ __align__(16) float tile[kTile];   // 3 planes x 10 rows x 128
  __shared__ float vmax[kTH * kW];              // vertical 3x3 (d,h) max
  __shared__ float red[256];

  const int tid = threadIdx.x;
  int bid = blockIdx.x;
  const int htile = bid % kHT; bid /= kHT;
  const int d = bid % kD;      bid /= kD;
  const int c = bid % kC;
  const int b = bid / kC;
  const int h0 = htile * kTH;

  const float NEG_INF = -__builtin_huge_valf();
  const size_t chan_base = (size_t)(b * kC + c) * kVol;

#if TDM_OK
  const bool interior =
      (d >= 1) && (d <= kD - 2) && (htile >= 1) && (htile <= kHT - 2);
#else
  const bool interior = false;
#endif

  if (!interior) {
    // Boundary path: every 16B chunk is either fully in-range (async DMA to
    // LDS) or a fully out-of-range halo row (-inf fill via ds_store).
    for (int i = tid; i < kVec; i += 256) {
      const int r = i >> 5;              // 0..29: plane-row index
      const int wq = (i & 31) << 2;      // w start, multiple of 4
      const int pp = r / 10;
      const int pr = r - pp * 10;
      const int dd = d + pp - 1;
      const int hh = h0 + pr - 1;
      float* ldst = &tile[r * kW + wq];
      if (dd >= 0 && dd < kD && hh >= 0 && hh < kH) {
        const float* gsrc = logits + chan_base + ((size_t)dd * kH + hh) * kW + wq;
#if USE_ASYNC_LDS
        const unsigned lds_addr = (unsigned)(uintptr_t)ldst;  // low 32b = LDS offset
        const unsigned long long gaddr = (unsigned long long)(uintptr_t)gsrc;
        asm volatile("global_load_async_to_lds_b128 %0, %1, off"
                     :: "v"(lds_addr), "v"(gaddr) : "memory");
#else
        const float4 v = *reinterpret_cast<const float4*>(gsrc);
        *reinterpret_cast<float4*>(ldst) = v;
#endif
      } else {
        float4 ninf;
        ninf.x = NEG_INF; ninf.y = NEG_INF; ninf.z = NEG_INF; ninf.w = NEG_INF;
        *reinterpret_cast<float4*>(ldst) = ninf;
      }
    }
#if USE_ASYNC_LDS
    asm volatile("s_wait_asynccnt 0" ::: "memory");
#endif
  }
#if TDM_OK
  else if (tid < 32) {  // wave 0 issues one TDM descriptor for the whole tile
    const size_t gbyte =
        (size_t)(uintptr_t)(logits + chan_base +
                            ((size_t)(d - 1) * kH + (h0 - 1)) * kW);
    const unsigned lds_base = (unsigned)(uintptr_t)&tile[0];
    // D# group 0: count=1 | lds_addr | global_addr[56:0] | type=2
    v4u g0;
    g0[0] = 1u;
    g0[1] = lds_base;
    g0[2] = (unsigned)(gbyte & 0xFFFFFFFFull);
    g0[3] = (unsigned)((gbyte >> 32) & 0x01FFFFFFull) | (2u << 30);
    // D# group 1: data_size=4B; tensor_dim0=W, tensor_dim1=H;
    // tile = 128 x 10 x 3; strides: row=W, plane=H*W (data_size units).
    v8i g1;
    g1[0] = 2 << 16;                       // data_size = 2 (4 bytes)
    g1[1] = (kW & 0xFFFF) << 16;           // tensor_dim0[15:0]
    g1[2] = (kH & 0xFFFF) << 16;           // tensor_dim0[31:16]=0 | tensor_dim1[15:0]
    g1[3] = kW << 16;                      // tensor_dim1[31:16]=0 | tile_dim0=128
    g1[4] = 10 | (3 << 16);                // tile_dim1=10, tile_dim2=3
    g1[5] = kW;                            // tensor_dim0_stride[31:0]
    g1[6] = (kH * kW) << 16;               // dim0_stride[47:32]=0 | dim1_stride[15:0]
    g1[7] = (kH * kW) >> 16;               // tensor_dim1_stride[47:16]
    // D# group 2/3: 3D tile -> tensor_dim2 only; rest zero.
    v4i g2 = {kD, 0, 0, 0};
    v4i g3 = {0, 0, 0, 0};
    // clang-23 / amdgpu-toolchain 6-arg form: extra int32x8 group (zero-filled
    // for a plain 3D, non-gather, non-cluster descriptor), then cpol.
    v8i g4 = {0, 0, 0, 0, 0, 0, 0, 0};
    __builtin_amdgcn_tensor_load_to_lds(g0, g1, g2, g3, g4, 0);
    __builtin_amdgcn_s_wait_tensorcnt(0);
  }
#endif
  __syncthreads();

  // Phase A: vertical max over 3 d-planes x 3 h-rows (9 LDS reads / voxel).
  for (int e = 0; e < 4; ++e) {
    const int o = tid + e * 256;      // 0..1023
    const int r = o >> 7;             // output row 0..7
    const int w = o & 127;
    float m = NEG_INF;
#pragma unroll
    for (int pp = 0; pp < 3; ++pp) {
#pragma unroll
      for (int rr = 0; rr < 3; ++rr) {
        m = fmaxf(m, tile[pp * kPlane + (r + rr) * kW + w]);
      }
    }
    vmax[o] = m;
  }
  __syncthreads();

  // Phase B: horizontal 3-max + mask-gated sigmoid focal terms.
  const int aidx = c % kA;  // channel c == cls*A + a
  const float* pg = prob_gt + (size_t)(b * kA + aidx) * kVol + (size_t)d * kH * kW;
  float lsum = 0.0f, csum = 0.0f;
  for (int e = 0; e < 4; ++e) {
    const int o = tid + e * 256;
    const int r = o >> 7;
    const int w = o & 127;
    float pooled = vmax[o];
    if (w > 0)      pooled = fmaxf(pooled, vmax[o - 1]);
    if (w < kW - 1) pooled = fmaxf(pooled, vmax[o + 1]);
    const float x = tile[kPlane + (r + 1) * kW + w];  // center value
    // Cheap mask first: negmask & NMS max & threshold (no transcendentals).
    const bool active = (pg[(h0 + r) * kW + w] == -1.0f) &&
                        (pooled == x) && (x > kXThresh);
    if (active) {  // ~1% lane density; ~70% of wave32s skip entirely
      const float s = 1.0f / (1.0f + expf(-x));
      const float p = fminf(fmaxf(s, kEPS), 1.0f - kEPS);
      const float wn = p * p;                           // p^ALPHA, ALPHA=2
      lsum = fmaf(-logf(1.0f - p), wn, lsum);
      csum += wn;
    }
  }

  // Deterministic block tree-reduction -> per-block partials.
  red[tid] = lsum; __syncthreads();
  for (int s2 = 128; s2 > 0; s2 >>= 1) {
    if (tid < s2) red[tid] += red[tid + s2];
    __syncthreads();
  }
  if (tid == 0) ws_loss[blockIdx.x] = red[0];
  __syncthreads();
  red[tid] = csum; __syncthreads();
  for (int s2 = 128; s2 > 0; s2 >>= 1) {
    if (tid < s2) red[tid] += red[tid + s2];
    __syncthreads();
  }
  if (tid == 0) ws_cnt[blockIdx.x] = red[0];
}

// ---------------------------------------------------------------------------
// Kernel 2: positive + "other-class" branch. One block per batch, 1 thread
// per coordinate record.
// ---------------------------------------------------------------------------
__global__ __launch_bounds__(kM) void focal_pos_kernel(
    const float* __restrict__ logits, const float* __restrict__ prob_gt,
    const int* __restrict__ coord, const float* __restrict__ wcls,
    float* __restrict__ pos_out /* [kB][4] = lpos, loth, cpos, coth */) {
  __shared__ int zrow[kNCLS];
  __shared__ float red[kM];
  const int b = blockIdx.x;
  const int m = threadIdx.x;
  if (m < kNCLS) zrow[m] = 0;
  __syncthreads();

  const int* cg = coord + (b * kM + m) * 4;
  const int c0 = cg[0];
  const bool valid = (c0 > -1);
  const int a  = valid ? c0    : 0;
  const int dd = valid ? cg[1] : 0;
  const int hh = valid ? cg[2] : 0;
  const int ww = valid ? cg[3] : 0;
  const size_t sp = ((size_t)dd * kH + hh) * kW + ww;
  const float pgv = prob_gt[(size_t)(b * kA + a) * kVol + sp];
  const int cls = valid ? ((int)pgv - 1) : 0;
  if (valid) atomicOr(&zrow[cls], 1);
  __syncthreads();

  float po[kNCLS];
#pragma unroll
  for (int ci = 0; ci < kNCLS; ++ci) {
    const float x = logits[(size_t)(b * kC + ci * kA + a) * kVol + sp];
    const float s = 1.0f / (1.0f + expf(-x));
    po[ci] = fminf(fmaxf(s, kEPS), 1.0f - kEPS);
  }
  const float vf = valid ? 1.0f : 0.0f;
  const float pt = po[cls];
  const float wpos = (1.0f - pt) * (1.0f - pt) * wcls[cls] * vf;
  const float lpos = -logf(pt) * wpos;
  const float cpos = wpos;
  float loth = 0.0f, coth = 0.0f;
#pragma unroll
  for (int ci = 0; ci < kNCLS; ++ci) {
    float wo = fmaxf(po[ci] - (pt - 0.1f), 0.0f);
    wo *= (po[ci] > 0.5f) ? 1.0f : 0.0f;
    wo *= (pt > 0.5f) ? 1.0f : 0.0f;
    wo = zrow[ci] ? 0.0f : wo;
    wo *= vf;
    loth = fmaf(-logf(1.0f - po[ci]), wo, loth);
    coth += wo;
  }

  float vals[4] = {lpos, loth, cpos, coth};
  for (int k = 0; k < 4; ++k) {
    __syncthreads();
    red[m] = vals[k];
    __syncthreads();
    for (int s2 = kM / 2; s2 > 0; s2 >>= 1) {
      if (m < s2) red[m] += red[m + s2];
      __syncthreads();
    }
    if (m == 0) pos_out[b * 4 + k] = red[0];
  }
}

// ---------------------------------------------------------------------------
// Kernel 3: deterministic final fold of all partials into the 6 outputs:
// [sum(l_pos), loss_neg, sum(l_oth), sum(c_pos), count_neg, sum(c_oth)]
// ---------------------------------------------------------------------------
__global__ __launch_bounds__(256) void focal_final_kernel(
    const float* __restrict__ ws_loss, const float* __restrict__ ws_cnt,
    const float* __restrict__ pos_out, float* __restrict__ out) {
  __shared__ float red[256];
  const int t = threadIdx.x;
  float l = 0.0f, cc = 0.0f;
  for (int i = t; i < kNBLK; i += 256) { l += ws_loss[i]; cc += ws_cnt[i]; }
  red[t] = l; __syncthreads();
  for (int s2 = 128; s2 > 0; s2 >>= 1) {
    if (t < s2) red[t] += red[t + s2];
    __syncthreads();
  }
  if (t == 0) out[1] = red[0];
  __syncthreads();
  red[t] = cc; __syncthreads();
  for (int s2 = 128; s2 > 0; s2 >>= 1) {
    if (t < s2) red[t] += red[t + s2];
    __syncthreads();
  }
  if (t == 0) {
    out[4] = red[0];
    out[0] = pos_out[0] + pos_out[4];
    out[2] = pos_out[1] + pos_out[5];
    out[3] = pos_out[2] + pos_out[6];
    out[5] = pos_out[3] + pos_out[7];
  }
}

// ---------------------------------------------------------------------------
extern "C" void kernel_launch(void* const* d_in, const int* in_sizes, int n_in,
                              void* d_out, int out_size, void* d_ws, size_t ws_size,
                              hipStream_t stream) {
  const float* logits  = (const float*)d_in[0];
  const float* prob_gt = (const float*)d_in[1];
  const int*   coord   = (const int*)d_in[2];
  const float* wcls    = (const float*)d_in[3];
  float* out = (float*)d_out;

  float* ws      = (float*)d_ws;
  float* ws_loss = ws;                 // kNBLK floats
  float* ws_cnt  = ws + kNBLK;         // kNBLK floats
  float* pos_out = ws + 2 * kNBLK;     // kB*4 floats

  focal_neg_kernel<<<kNBLK, 256, 0, stream>>>(logits, prob_gt, ws_loss, ws_cnt);
  focal_pos_kernel<<<kB, kM, 0, stream>>>(logits, prob_gt, coord, wcls, pos_out);
  focal_final_kernel<<<1, 256, 0, stream>>>(ws_loss, ws_cnt, pos_out, out);
}